// Router_68547678044792
// MI455X (gfx1250) — compile-verified
//
#include <hip/hip_runtime.h>
#include <math.h>

typedef __attribute__((ext_vector_type(2))) float    v2f;
typedef __attribute__((ext_vector_type(4))) float    v4f;
typedef __attribute__((ext_vector_type(8))) float    v8f;
typedef __attribute__((ext_vector_type(4))) int      v4i;
typedef __attribute__((ext_vector_type(8))) int      v8i;
typedef __attribute__((ext_vector_type(4))) unsigned v4u;

#define D_DIM 4096      // K (hidden dim)
#define E_DIM 128       // N (experts)
#define TOPK 8
#define M_TILE 64       // tokens per workgroup
#define KC 64           // K-chunk staged in LDS
#define XSTRIDE 68      // padded row stride (floats): 64 data + 4 pad (TDM pad reproduces this)
#define XBUF (M_TILE * XSTRIDE)       // 4352 floats per buffer
#define LG_STRIDE 132   // logits row stride (overlay: 64*132 <= 2*XBUF)
#define NCHUNK (D_DIM / KC)           // 64

#if defined(__gfx1250__) && __has_builtin(__builtin_amdgcn_tensor_load_to_lds)
#define HAVE_TDM 1
#else
#define HAVE_TDM 0
#endif

#if defined(__gfx1250__) && __has_builtin(__builtin_amdgcn_global_load_async_to_lds_b128)
#define HAVE_ASYNC_LDS 1
#else
#define HAVE_ASYNC_LDS 0
#endif

#if HAVE_TDM || HAVE_ASYNC_LDS
typedef __attribute__((address_space(1))) v4i*  gptr128;
typedef __attribute__((address_space(3))) v4i*  lptr128;
typedef __attribute__((address_space(3))) float* lptrf;
#endif

// ---------------- TDM staging: one descriptor moves the whole 64x64 f32 tile ----------------
#if HAVE_TDM
__device__ __forceinline__ void stage_x_tdm(const float* x, float* ldsbuf, int tok0, int k0) {
    // D#.lds_addr is a byte offset within the workgroup's LDS allocation.
    unsigned lds_addr = (unsigned)(size_t)(lptrf)ldsbuf;
    unsigned long long ga = (unsigned long long)(size_t)(x + (size_t)tok0 * D_DIM + k0);

    // Group 0: count=1 | lds_addr | global_addr[56:0] | type=2
    v4u g0;
    g0[0] = 1u;                                   // count=1, user descriptor
    g0[1] = lds_addr;                             // bits 63:32
    g0[2] = (unsigned)(ga & 0xFFFFFFFFu);         // global_addr[31:0]
    g0[3] = (unsigned)((ga >> 32) & 0x1FFFFFFu)   // global_addr[56:32]
          | (2u << 30);                           // type = 2 ("image")

    // Group 1: data_size=4B, pad 4 DWORDs every 64 DWORDs, 2D tile 64x64,
    //          tensor_dim0=4096, tensor_dim1=16384, dim0 stride=4096 elements.
    v8i g1;
    g1[0] = (2 << 16)        // data_size = 2 -> 4 bytes
          | (1 << 20)        // pad_enable (load only)
          | (5 << 22)        // pad_interval: 1<<(5+1) = 64 DWORDs between pads
          | (3 << 25);       // pad_amount: 3+1 = 4 DWORDs -> row stride 68 floats
    g1[1] = (int)((D_DIM & 0xFFFF) << 16);        // tensor_dim0[15:0]   (bits 63:48)
    g1[2] = (int)(((D_DIM >> 16) & 0xFFFF)        // tensor_dim0[31:16]  (bits 79:64)
          | ((16384 & 0xFFFF) << 16));            // tensor_dim1[15:0]   (bits 95:80)
    g1[3] = (int)(((16384 >> 16) & 0xFFFF)        // tensor_dim1[31:16]
          | (KC << 16));                          // tile_dim0 = 64      (bits 127:112)
    g1[4] = (int)(M_TILE & 0xFFFF);               // tile_dim1 = 64; tile_dim2 = 0
    g1[5] = (int)D_DIM;                           // tensor_dim0_stride[31:0] = 4096
    g1[6] = 0;                                    // stride0 hi | tensor_dim1_stride lo
    g1[7] = 0;                                    // tensor_dim1_stride hi

    v4i gz  = (v4i)0;                             // groups 2/3 unused (2D tensor)
    v8i gz8 = (v8i)0;                             // 5th group arg (clang-23 6-arg form)
    __builtin_amdgcn_tensor_load_to_lds(g0, g1, gz, gz, gz8, /*cpol=*/0);
}
#endif

// ---------------- per-lane async fallback staging ----------------
template <int N>
__device__ __forceinline__ void wait_async() {
#if !HAVE_TDM && HAVE_ASYNC_LDS
#if __has_builtin(__builtin_amdgcn_s_wait_asynccnt)
    __builtin_amdgcn_s_wait_asynccnt(N);
#else
    asm volatile("s_wait_asynccnt %0" ::"i"(N) : "memory");
#endif
#endif
}

__device__ __forceinline__ void stage_x(const float* __restrict__ x, float* dst,
                                        int tok0, int k0, int tid) {
#if HAVE_ASYNC_LDS
    #pragma unroll
    for (int i = 0; i < 4; ++i) {
        int idx = i * 256 + tid;      // float4 index; 16 float4 per row
        int row = idx >> 4;
        int c4  = idx & 15;
        const float* g = x + (size_t)(tok0 + row) * D_DIM + k0 + c4 * 4;
        float* l = dst + row * XSTRIDE + c4 * 4;
        __builtin_amdgcn_global_load_async_to_lds_b128((gptr128)g, (lptr128)l, 0, 0);
    }
#else
    v4f t[4];
    #pragma unroll
    for (int i = 0; i < 4; ++i) {
        int idx = i * 256 + tid; int row = idx >> 4, c4 = idx & 15;
        t[i] = __builtin_nontemporal_load((const v4f*)(x + (size_t)(tok0 + row) * D_DIM + k0 + c4 * 4));
    }
    #pragma unroll
    for (int i = 0; i < 4; ++i) {
        int idx = i * 256 + tid;
        *(v4f*)&dst[(idx >> 4) * XSTRIDE + (idx & 15) * 4] = t[i];
    }
#endif
}

__global__ __launch_bounds__(256) void router_gemm_softmax_topk(
    const float* __restrict__ x, const float* __restrict__ W,
    const float* __restrict__ b, float* __restrict__ out_scores,
    int* __restrict__ out_idx, int n_tokens)
{
    __shared__ float lds[2 * XBUF];   // 34,816 bytes; logits phase overlays this

    const int tid  = threadIdx.x;
    const int wave = tid >> 5;        // 0..7 -> 16-expert N tile
    const int lane = tid & 31;
    const int tok0 = blockIdx.x * M_TILE;

    const int e0   = wave * 16;
    const int nIdx = lane & 15;        // M index (A frag) / N index (B,C,D)
    const int kb   = (lane >> 4) * 2;  // lanes 0-15: K 0/1, lanes 16-31: K 2/3

    v8f acc[4];
    #pragma unroll
    for (int mt = 0; mt < 4; ++mt) acc[mt] = (v8f)0.0f;

    const float* Wrow = W + (size_t)(e0 + nIdx) * D_DIM;

    // Prologue: kick off chunk 0 into buffer 0
#if HAVE_TDM
    if (tid < 32) stage_x_tdm(x, &lds[0], tok0, 0);
#else
    stage_x(x, &lds[0], tok0, 0, tid);
#endif

    int cur = 0;
    for (int c = 0; c < NCHUNK; ++c) {
#if HAVE_TDM
        if (tid < 32) {   // wave 0 drives the DMA; TENSORcnt is per-wave and in order
            if (c + 1 < NCHUNK) {
                stage_x_tdm(x, &lds[(cur ^ 1) * XBUF], tok0, (c + 1) * KC);
                __builtin_amdgcn_s_wait_tensorcnt(1);  // <=1 outstanding => chunk c resident
            } else {
                __builtin_amdgcn_s_wait_tensorcnt(0);
            }
        }
#else
        if (c + 1 < NCHUNK) {
            stage_x(x, &lds[(cur ^ 1) * XBUF], tok0, (c + 1) * KC, tid);
            wait_async<4>();
        } else {
            wait_async<0>();
        }
#endif
        __syncthreads();   // current buffer visible to all waves

        const float* xa = &lds[cur * XBUF];
        const float* Wk = Wrow + c * KC;
        for (int kk = 0; kk < KC; kk += 4) {
            v2f bfrag = *(const v2f*)(Wk + kk + kb);            // B[k][n] = W[e0+n][k]
            #pragma unroll
            for (int mt = 0; mt < 4; ++mt) {
                v2f afrag = *(const v2f*)&xa[(mt * 16 + nIdx) * XSTRIDE + kk + kb];
                acc[mt] = __builtin_amdgcn_wmma_f32_16x16x4_f32(
                    false, afrag, false, bfrag, (short)0, acc[mt], false, false);
            }
        }
        __syncthreads();   // done reading before this buffer is re-filled
        cur ^= 1;
    }

    // ---- scatter logits (+bias) into LDS per C/D 16x16 f32 layout ----
    float* logits = lds;
    const float bias = b[e0 + nIdx];
    const int mOff = (lane >> 4) * 8;  // lanes 16-31 hold M = r+8
    #pragma unroll
    for (int mt = 0; mt < 4; ++mt) {
        #pragma unroll
        for (int r = 0; r < 8; ++r) {
            int token = mt * 16 + r + mOff;
            logits[token * LG_STRIDE + e0 + nIdx] = acc[mt][r] + bias;
        }
    }
    __syncthreads();

    // ---- softmax + top-8 (one thread per token; negligible vs GEMM) ----
    if (tid < M_TILE) {
        const float* lg = &logits[tid * LG_STRIDE];
        float mx = lg[0];
        for (int e = 1; e < E_DIM; ++e) mx = fmaxf(mx, lg[e]);
        float sum = 0.0f;
        for (int e = 0; e < E_DIM; ++e) sum += __expf(lg[e] - mx);
        float inv = 1.0f / sum;

        unsigned long long takenLo = 0ull, takenHi = 0ull;
        const size_t gtok = (size_t)(tok0 + tid);
        for (int j = 0; j < TOPK; ++j) {
            float best = -INFINITY;
            int bi = 0;
            for (int e = 0; e < E_DIM; ++e) {
                bool used = (e < 64) ? ((takenLo >> e) & 1ull)
                                     : ((takenHi >> (e - 64)) & 1ull);
                float v = lg[e];
                if (!used && v > best) { best = v; bi = e; }  // strict > : first index wins ties
            }
            if (bi < 64) takenLo |= (1ull << bi); else takenHi |= (1ull << (bi - 64));
            out_scores[gtok * TOPK + j] = __expf(best - mx) * inv;
            out_idx[gtok * TOPK + j]    = bi;
        }
    }
}

extern "C" void kernel_launch(void* const* d_in, const int* in_sizes, int n_in,
                              void* d_out, int out_size, void* d_ws, size_t ws_size,
                              hipStream_t stream) {
    const float* x = (const float*)d_in[0];   // [B,S,D] f32
    const float* W = (const float*)d_in[1];   // [E,D]   f32
    const float* b = (const float*)d_in[2];   // [E]     f32
    // d_in[3] = top_k (== 8, compiled in)

    const int n_tokens = in_sizes[0] / D_DIM;             // B*S = 16384
    float* out_scores  = (float*)d_out;                   // [n_tokens, 8] f32
    int*   out_idx     = (int*)((float*)d_out + (size_t)n_tokens * TOPK); // [n_tokens, 8] i32

    const int grid = n_tokens / M_TILE;                   // 256 workgroups
    router_gemm_softmax_topk<<<grid, 256, 0, stream>>>(x, W, b, out_scores, out_idx, n_tokens);
}